// GraphOutputLayerWithPruning_46651934769540
// MI455X (gfx1250) — compile-verified
//
#include <hip/hip_runtime.h>
#include <hip/hip_bf16.h>
#include <math.h>

typedef float v2f __attribute__((ext_vector_type(2)));
typedef float v8f __attribute__((ext_vector_type(8)));

#define HDIM  512
#define NHEAD 8
#define DKDIM 64

// ---------------------------------------------------------------------------
// CDNA5 async global->LDS copies (ASYNCcnt tracked), via inline asm so the
// source is portable across ROCm-7.2 / amdgpu-toolchain builtin differences.
// ---------------------------------------------------------------------------
__device__ __forceinline__ unsigned lds_off_u32(const void* p) {
    // generic pointer to LDS: low 32 bits carry the LDS byte offset
    return (unsigned)(unsigned long long)(uintptr_t)p;
}
__device__ __forceinline__ void async_g2l_b128(unsigned lds, const void* g) {
    asm volatile("global_load_async_to_lds_b128 %0, %1, off"
                 :: "v"(lds), "v"((unsigned long long)(uintptr_t)g) : "memory");
}
__device__ __forceinline__ void wait_async0() {
    asm volatile("s_wait_asynccnt 0x0" ::: "memory");
}

// ---------------------------------------------------------------------------
// Phase 0: outputs = inputs (mask all-true), mask -> 1.0f, loss slot -> 0
// ---------------------------------------------------------------------------
__global__ void copy_outputs_kernel(const float* __restrict__ in,
                                    float* __restrict__ out,
                                    int n_f4, int mask_n,
                                    float* __restrict__ loss_slot) {
    int t = blockIdx.x * blockDim.x + threadIdx.x;
    if (t < n_f4) {
        ((float4*)out)[t] = ((const float4*)in)[t];
    }
    if (t < mask_n) out[(size_t)n_f4 * 4 + t] = 1.0f;
    if (t == 0) *loss_slot = 0.0f;
}

__global__ void zero_kernel(float* __restrict__ p, int n) {
    int t = blockIdx.x * blockDim.x + threadIdx.x;
    if (t < n) p[t] = 0.0f;
}

// ---------------------------------------------------------------------------
// fp32 WMMA GEMM, K = 512 fixed.
//   Y[:, ncb:ncb+256] = f( gather(X) @ W + bias ),  block tile 16 x 256,
//   grid = (nrows/16, Nout/256), block = 256 threads (8 waves, 2 tiles each).
//
// Double-buffered 16-K chunks, both operands DMA'd via
// GLOBAL_LOAD_ASYNC_TO_LDS_B128 (chunk c+1 copies while chunk c computes):
//   A chunk  As[2][16][20]  : pitch 20 (== 4 mod 16) -> A-fragment
//                             {A[m][ka],A[m][ka+1]} is one conflict-free b64;
//                             the K+2 half-wave lands in the bank holes.
//   B chunk  Bs[2][16][264] : 2*264 == 16 mod 64 -> the two half-waves read
//                             disjoint bank ranges on each b32 fragment load.
// All steady-state global/LDS addresses are induction pointers (one
// v_add_nc_u64 per stream per chunk -- no per-chunk 64-bit multiplies).
// Inner loop: 4 k-steps x 2 tiles V_WMMA_F32_16X16X4_F32, branchless,
// two named accumulators.
// ZDIV variant (o = wv / z) stages A manually (divide fused), B stays async.
// ---------------------------------------------------------------------------
template <bool TANH, bool ZDIV>
__global__ __launch_bounds__(256)
void wmma_gemm_k512(const float* __restrict__ X, const int* __restrict__ gidx,
                    const float* __restrict__ zdiv,
                    const float* __restrict__ W, const float* __restrict__ bias,
                    float* __restrict__ Y, int Nout) {
    __shared__ float As[2 * 16 * 20];    //  2,560 B
    __shared__ float Bs[2 * 16 * 264];   // 33,792 B

    const int tid  = threadIdx.x;
    const int row0 = blockIdx.x * 16;
    const int ncb  = blockIdx.y * 256;

    // per-thread staging coordinates
    const int bkr0 = tid >> 6;           // B: K-row for i=0 (i adds 4)
    const int bc4  = tid & 63;           // B: float4 column
    const int ar   = tid >> 2;           // A async: row (threads 0..63)
    const int ac4  = tid & 3;            // A async: float4 within 16-K chunk
    const int mr   = tid >> 4;           // A manual (ZDIV): row
    const int mc   = tid & 15;           // A manual (ZDIV): K within chunk

    const int arow = ZDIV ? mr : ar;
    int srow = row0 + arow;
    if (gidx) srow = gidx[srow];
    const float* Xrow = X + (size_t)srow * HDIM;

    // ---- induction pointers (advance by one 64-bit add per chunk) ----
    const size_t wadv = (size_t)16 * Nout;            // floats per chunk
    const float* wpa = W + ncb + (size_t)bkr0 * Nout + bc4 * 4;
    const float* wpb = wpa + (size_t)4 * Nout;
    const float* wpc = wpa + (size_t)8 * Nout;
    const float* wpd = wpa + (size_t)12 * Nout;
    const float* apg = Xrow + ac4 * 4;                // async A stream
    const float* apm = Xrow + mc;                     // manual A stream (ZDIV)

    // LDS byte offsets, both buffers, precomputed
    const unsigned blB0 = lds_off_u32(Bs) + (unsigned)((bkr0 * 264 + bc4 * 4) * 4);
    const unsigned blB1 = blB0 + 16 * 264 * 4;
    const unsigned blA0 = lds_off_u32(As) + (unsigned)((ar * 20 + ac4 * 4) * 4);
    const unsigned blA1 = blA0 + 16 * 20 * 4;
    const unsigned BROW4 = 264 * 4 * 4;               // 4 K-rows of B in bytes

    // ---- issue chunk 0 into buffer 0 ----
    {
        async_g2l_b128(blB0,             wpa);
        async_g2l_b128(blB0 + BROW4,     wpb);
        async_g2l_b128(blB0 + BROW4 * 2, wpc);
        async_g2l_b128(blB0 + BROW4 * 3, wpd);
        wpa += wadv; wpb += wadv; wpc += wadv; wpd += wadv;
        if (ZDIV) {
            float inv = 1.0f / zdiv[(size_t)(row0 + mr) * NHEAD + 0];
            As[mr * 20 + mc] = apm[0] * inv;
            apm += 16;
        } else if (tid < 64) {
            async_g2l_b128(blA0, apg);
            apg += 16;
        }
    }
    wait_async0();
    __syncthreads();

    const int lane = tid & 31;
    const int wave = tid >> 5;
    const int hi   = lane >> 4;          // half-wave: K +0 / +2
    const int lo   = lane & 15;          // A row / B,C,D column in tile
    const int nc0  = wave * 32 + lo;     // local col of tile 0 (tile 1: +16)

    v8f acc0 = {};
    v8f acc1 = {};

    #pragma unroll 2
    for (int c = 0; c < 32; ++c) {
        const float* Ab = &As[(c & 1) * (16 * 20)];
        const float* Bb = &Bs[(c & 1) * (16 * 264)];

        // ---- issue chunk c+1 into the other buffer (overlaps compute) ----
        if (c + 1 < 32) {
            const unsigned blB = ((c + 1) & 1) ? blB1 : blB0;
            async_g2l_b128(blB,             wpa);
            async_g2l_b128(blB + BROW4,     wpb);
            async_g2l_b128(blB + BROW4 * 2, wpc);
            async_g2l_b128(blB + BROW4 * 3, wpd);
            wpa += wadv; wpb += wadv; wpc += wadv; wpd += wadv;
            if (ZDIV) {
                int kk = (c + 1) * 16 + mc;
                float inv = 1.0f / zdiv[(size_t)(row0 + mr) * NHEAD + (kk >> 6)];
                float* An = &As[((c + 1) & 1) * (16 * 20)];
                An[mr * 20 + mc] = apm[0] * inv;
                apm += 16;
            } else if (tid < 64) {
                async_g2l_b128(((c + 1) & 1) ? blA1 : blA0, apg);
                apg += 16;
            }
        }

        // ---- compute chunk c: 4 k-steps x 2 tiles ----
        #pragma unroll
        for (int ks = 0; ks < 4; ++ks) {
            int kl = ks * 4 + 2 * hi;                 // local K (even)
            v2f a  = *(const v2f*)&Ab[lo * 20 + kl];
            v2f b0, b1;
            b0.x = Bb[kl * 264 + nc0];
            b0.y = Bb[(kl + 1) * 264 + nc0];
            b1.x = Bb[kl * 264 + nc0 + 16];
            b1.y = Bb[(kl + 1) * 264 + nc0 + 16];
            acc0 = __builtin_amdgcn_wmma_f32_16x16x4_f32(
                false, a, false, b0, (short)0, acc0, false, false);
            acc1 = __builtin_amdgcn_wmma_f32_16x16x4_f32(
                false, a, false, b1, (short)0, acc1, false, false);
        }

        wait_async0();        // chunk c+1 landed (no-op on last iteration)
        __syncthreads();      // all waves done with buffers
    }

    // ---- epilogue: C/D layout row = r + 8*hi, col = lo ----
    {
        int col0 = ncb + wave * 32 + lo;
        int col1 = col0 + 16;
        float b0 = bias ? bias[col0] : 0.0f;
        float b1 = bias ? bias[col1] : 0.0f;
        #pragma unroll
        for (int r = 0; r < 8; ++r) {
            int mrow = row0 + r + 8 * hi;
            float v0 = acc0[r] + b0;
            float v1 = acc1[r] + b1;
            if (TANH) { v0 = tanhf(v0); v1 = tanhf(v1); }
            Y[(size_t)mrow * Nout + col0] = v0;
            Y[(size_t)mrow * Nout + col1] = v1;
        }
    }
}

// ---------------------------------------------------------------------------
// Edge phase: thread per (edge, head).
//   score = exp(clamp(dot(k[src,h], q[dst,h]) / 8, -5, 5))
//   z[dst,h]    += score                 (atomic)
//   wv[dst,h,:] += v[src,h,:] * score    (atomic)
// k/q/v (200 MB) are L2-resident (192 MB L2) -> gather traffic stays on-chip.
// ---------------------------------------------------------------------------
__global__ void edge_attn_kernel(const float* __restrict__ k,
                                 const float* __restrict__ q,
                                 const float* __restrict__ v,
                                 const int* __restrict__ esrc,
                                 const int* __restrict__ edst,
                                 float* __restrict__ wv,
                                 float* __restrict__ z, int nwork) {
    int gid = blockIdx.x * blockDim.x + threadIdx.x;
    if (gid >= nwork) return;
    int e = gid >> 3, h = gid & 7;
    int s = esrc[e], d = edst[e];
    const float* kp = k + (size_t)s * HDIM + h * DKDIM;
    const float* qp = q + (size_t)d * HDIM + h * DKDIM;
    float dot = 0.0f;
    #pragma unroll
    for (int i = 0; i < DKDIM; i += 4) {
        float4 kv = *(const float4*)(kp + i);
        float4 qv = *(const float4*)(qp + i);
        dot += kv.x * qv.x + kv.y * qv.y + kv.z * qv.z + kv.w * qv.w;
    }
    float sc = __expf(fminf(fmaxf(dot * 0.125f, -5.0f), 5.0f));
    atomicAdd(&z[(size_t)d * NHEAD + h], sc);
    const float* vp = v + (size_t)s * HDIM + h * DKDIM;
    float* wvp = wv + (size_t)d * HDIM + h * DKDIM;
    #pragma unroll 4
    for (int i = 0; i < DKDIM; ++i) atomicAdd(&wvp[i], vp[i] * sc);
}

// ---------------------------------------------------------------------------
// logits + BCEWithLogits(sum). One wave32 per schema node.
// ---------------------------------------------------------------------------
__global__ void loss_kernel(const float* __restrict__ cq,
                            const float* __restrict__ cs,
                            const float* __restrict__ Waff,
                            const float* __restrict__ baff,
                            const float* __restrict__ label,
                            float* __restrict__ loss_out, int ns) {
    int node = blockIdx.x * 8 + (threadIdx.x >> 5);
    int lane = threadIdx.x & 31;
    if (node >= ns) return;
    const int HC = HDIM / 2;
    float acc = 0.0f;
    for (int i = lane; i < HC; i += 32) {
        acc += cq[(size_t)node * HC + i] * Waff[i]
             + cs[(size_t)node * HC + i] * Waff[HC + i];
    }
    #pragma unroll
    for (int off = 16; off > 0; off >>= 1) acc += __shfl_down(acc, off, 32);
    if (lane == 0) {
        float l = acc + baff[0];
        float y = label[node];
        float li = fmaxf(l, 0.0f) - l * y + log1pf(__expf(-fabsf(l)));
        atomicAdd(loss_out, li);
    }
}

// ---------------------------------------------------------------------------
extern "C" void kernel_launch(void* const* d_in, const int* in_sizes, int n_in,
                              void* d_out, int out_size, void* d_ws, size_t ws_size,
                              hipStream_t stream) {
    const float* inputs = (const float*)d_in[0];
    const int*   qidx   = (const int*)d_in[2];
    const int*   sidx   = (const int*)d_in[3];
    const int*   esrc   = (const int*)d_in[4];
    const int*   edst   = (const int*)d_in[5];
    const float* label  = (const float*)d_in[6];
    const float* Wq  = (const float*)d_in[7];
    const float* bq  = (const float*)d_in[8];
    const float* Wk  = (const float*)d_in[9];
    const float* Wv  = (const float*)d_in[10];
    const float* Wo  = (const float*)d_in[11];
    const float* bo  = (const float*)d_in[12];
    const float* Wmq = (const float*)d_in[13];
    const float* bmq = (const float*)d_in[14];
    const float* Wms = (const float*)d_in[15];
    const float* bms = (const float*)d_in[16];
    const float* Waff = (const float*)d_in[17];
    const float* baff = (const float*)d_in[18];

    const int NQ = in_sizes[2];
    const int NS = in_sizes[3];
    const int E  = in_sizes[4];
    const int NTOT = in_sizes[0] / HDIM;   // B*L

    float* out = (float*)d_out;
    float* loss_slot = out + (size_t)NTOT * HDIM + NTOT;

    // workspace layout (floats)
    float* ws = (float*)d_ws;
    size_t o = 0;
    float* q  = ws + o; o += (size_t)NS * HDIM;
    float* k  = ws + o; o += (size_t)NQ * HDIM;
    float* v  = ws + o; o += (size_t)NQ * HDIM;
    float* wv = ws + o; o += (size_t)NS * HDIM;     // wv,z contiguous -> one zero pass
    float* z  = ws + o; o += (size_t)NS * NHEAD;
    float* nc = ws + o; o += (size_t)NS * HDIM;
    float* cq = ws + o; o += (size_t)NS * (HDIM / 2);
    float* cs = ws + o; o += (size_t)NS * (HDIM / 2);

    // 0) outputs copy + mask + loss init
    {
        int n_f4 = NTOT * HDIM / 4;
        copy_outputs_kernel<<<(n_f4 + 255) / 256, 256, 0, stream>>>(
            inputs, out, n_f4, NTOT, loss_slot);
    }
    // zero wv + z
    {
        int n = NS * HDIM + NS * NHEAD;
        zero_kernel<<<(n + 255) / 256, 256, 0, stream>>>(wv, n);
    }
    // 1) q = schema @ Wq + bq ; k = question @ Wk ; v = question @ Wv
    wmma_gemm_k512<false, false><<<dim3(NS / 16, 2), 256, 0, stream>>>(
        inputs, sidx, nullptr, Wq, bq, q, HDIM);
    wmma_gemm_k512<false, false><<<dim3(NQ / 16, 2), 256, 0, stream>>>(
        inputs, qidx, nullptr, Wk, nullptr, k, HDIM);
    wmma_gemm_k512<false, false><<<dim3(NQ / 16, 2), 256, 0, stream>>>(
        inputs, qidx, nullptr, Wv, nullptr, v, HDIM);
    // 2) edge scores + scatter aggregation
    {
        int nwork = E * NHEAD;
        edge_attn_kernel<<<(nwork + 255) / 256, 256, 0, stream>>>(
            k, q, v, esrc, edst, wv, z, nwork);
    }
    // 3) node_context = (wv / z) @ Wo + bo   (division fused into A staging)
    wmma_gemm_k512<false, true><<<dim3(NS / 16, 2), 256, 0, stream>>>(
        wv, nullptr, z, Wo, bo, nc, HDIM);
    // 4) cq = tanh(nc @ Wmq + bmq) ; cs = tanh(schema @ Wms + bms)
    wmma_gemm_k512<true, false><<<dim3(NS / 16, 1), 256, 0, stream>>>(
        nc, nullptr, nullptr, Wmq, bmq, cq, HDIM / 2);
    wmma_gemm_k512<true, false><<<dim3(NS / 16, 1), 256, 0, stream>>>(
        inputs, sidx, nullptr, Wms, bms, cs, HDIM / 2);
    // 5) logits + BCE sum -> loss slot
    loss_kernel<<<(NS + 7) / 8, 256, 0, stream>>>(cq, cs, Waff, baff, label,
                                                  loss_slot, NS);
}